// AtLocCriterion_10857677324673
// MI455X (gfx1250) — compile-verified
//
#include <hip/hip_runtime.h>
#include <stdint.h>

#define TPB 256
#define IPB 256                    // batch items per block
#define FPB (IPB * 54)             // floats per tar tile (13824)
#define EPS4F 8.881784197001252e-16f

typedef __attribute__((ext_vector_type(2))) float v2f;
typedef __attribute__((ext_vector_type(8))) float v8f;

// ---- CDNA5 async global->LDS copy (GVS addressing: saddr base + 32-bit voffset)
__device__ __forceinline__ void async_lds_b128(uint32_t lds_addr, uint32_t voff, const void* sbase) {
  asm volatile("global_load_async_to_lds_b128 %0, %1, %2"
               :: "v"(lds_addr), "v"(voff), "s"(sbase) : "memory");
}
__device__ __forceinline__ void async_lds_b32(uint32_t lds_addr, uint32_t voff, const void* sbase) {
  asm volatile("global_load_async_to_lds_b32 %0, %1, %2"
               :: "v"(lds_addr), "v"(voff), "s"(sbase) : "memory");
}
__device__ __forceinline__ void wait_async_zero() {
  asm volatile("s_wait_asynccnt 0" ::: "memory");
}
// flat pointer to __shared__ -> LDS byte address (LDS aperture passes addr[31:0] through)
__device__ __forceinline__ uint32_t lds_addr_of(const void* p) {
  return (uint32_t)(uintptr_t)p;
}

// R = xmat @ ymat @ zmat (reference euler2mat), T = vec[0:3]
__device__ __forceinline__ void pose_from_vec(const float* __restrict__ v,
                                              float R[3][3], float T[3]) {
  T[0] = v[0]; T[1] = v[1]; T[2] = v[2];
  const float x = v[3], y = v[4], z = v[5];
  const float cx = cosf(x), sx = sinf(x);
  const float cy = cosf(y), sy = sinf(y);
  const float cz = cosf(z), sz = sinf(z);
  R[0][0] = cy * cz;               R[0][1] = -cy * sz;              R[0][2] = sy;
  R[1][0] = cx * sz + sx * sy * cz; R[1][1] = cx * cz - sx * sy * sz; R[1][2] = -sx * cy;
  R[2][0] = sx * sz - cx * sy * cz; R[2][1] = sx * cz + cx * sy * sz; R[2][2] = cx * cy;
}

__global__ __launch_bounds__(TPB) void crit_main(const float* __restrict__ pred,
                                                 const float* __restrict__ tar,
                                                 const float* __restrict__ glp,
                                                 float* __restrict__ partA,
                                                 float* __restrict__ partB,
                                                 int nBatch) {
  __shared__ __align__(16) float smem[FPB];
  __shared__ float wredA[16], wredB[16];   // 8 waves x 2 halves
  const int tid = threadIdx.x;
  const int blk = blockIdx.x;
  const size_t e0 = (size_t)blk * FPB;
  const size_t totalF = (size_t)nBatch * 54;
  const int nf = (int)((totalF - e0 < (size_t)FPB) ? (totalF - e0) : (size_t)FPB);
  const int nv4 = nf >> 2;

  // ---- Stage this block's tar tile into LDS via async-tensor path (no VGPR staging)
  const char* tbase = (const char*)(tar + e0);
  for (int i = tid; i < nv4; i += TPB)
    async_lds_b128(lds_addr_of(&smem[i * 4]), (uint32_t)(i * 16), tbase);
  for (int i = (nv4 << 2) + tid; i < nf; i += TPB)
    async_lds_b32(lds_addr_of(&smem[i]), (uint32_t)(i * 4), tbase);
  wait_async_zero();   // this wave's async copies landed in LDS
  __syncthreads();     // all waves' copies visible

  // ---- Term 1: streaming sum |pred - targ| (b128 coalesced pred, LDS tar)
  float acc1 = 0.f;
  {
    const float4* __restrict__ p4 = (const float4*)(pred + e0);
    const float4* s4 = (const float4*)smem;
    for (int i = tid; i < nv4; i += TPB) {
      float4 p = p4[i];
      float4 t = s4[i];
      acc1 += fabsf(p.x - t.x) + fabsf(p.y - t.y) + fabsf(p.z - t.z) + fabsf(p.w - t.w);
    }
    for (int i = (nv4 << 2) + tid; i < nf; i += TPB)
      acc1 += fabsf(pred[e0 + i] - smem[i]);
  }

  // ---- Term 2: 9-pose affine chain + mat2euler, one batch item per thread
  float acc2 = 0.f;
  const int b = blk * IPB + tid;
  if (b < nBatch) {
    const float* tv = smem + tid * 54;  // conflict-free: 54 = -10 mod 64 banks
    float R[3][3], T[3];
    pose_from_vec(tv, R, T);
    for (int i = 1; i < 9; ++i) {
      float Ri[3][3], Ti[3], nR[3][3], nT[3];
      pose_from_vec(tv + i * 6, Ri, Ti);
#pragma unroll
      for (int r = 0; r < 3; ++r) {
#pragma unroll
        for (int c = 0; c < 3; ++c)
          nR[r][c] = R[r][0] * Ri[0][c] + R[r][1] * Ri[1][c] + R[r][2] * Ri[2][c];
        nT[r] = R[r][0] * Ti[0] + R[r][1] * Ti[1] + R[r][2] * Ti[2] + T[r];
      }
#pragma unroll
      for (int r = 0; r < 3; ++r) {
#pragma unroll
        for (int c = 0; c < 3; ++c) R[r][c] = nR[r][c];
        T[r] = nT[r];
      }
    }
    const float cy2 = sqrtf(R[0][0] * R[0][0] + R[1][0] * R[1][0]);
    const bool cond = cy2 > EPS4F;
    const float ax = cond ? atan2f(R[2][1], R[2][2]) : atan2f(-R[1][2], R[1][1]);
    const float ay = atan2f(-R[2][0], cy2);
    const float az = cond ? atan2f(R[1][0], R[0][0]) : 1.0f;
    const float q[6] = {T[0], T[1], T[2], ax, ay, az};
    const float* g = glp + (size_t)b * 6;
#pragma unroll
    for (int k = 0; k < 6; ++k) acc2 += fabsf(q[k] - g[k]);
  }

  // ---- WMMA wave reduction: one V_WMMA_F32_16X16X4_F32 reduces both sums
  // A (16x4, lane<->M, vgpr+half<->K): a.x=acc1, a.y=acc2
  // B (4x16, lane<->N, vgpr+half<->K): cols n<8 select acc1 (K=0,2), n>=8 select acc2 (K=1,3)
  // C[m][n<8]  = acc1(m)+acc1(16+m);  C[m][n>=8] = acc2(m)+acc2(16+m)
  {
    const unsigned lane = (unsigned)tid & 31u;
    const unsigned wave = (unsigned)tid >> 5;
    const unsigned n = lane & 15u;
    v2f a; a.x = acc1; a.y = acc2;
    v2f bsel; bsel.x = (n < 8u) ? 1.f : 0.f; bsel.y = (n < 8u) ? 0.f : 1.f;
    v8f c = {};
    c = __builtin_amdgcn_wmma_f32_16x16x4_f32(false, a, false, bsel,
                                              (short)0, c, false, false);
    // in-lane fold of the 8 C rows (no cross-lane ops needed)
    const float s = ((c[0] + c[1]) + (c[2] + c[3])) + ((c[4] + c[5]) + (c[6] + c[7]));
    // lanes 0 / 16 carry the two acc1 halves; lanes 8 / 24 the two acc2 halves
    if (lane == 0u || lane == 16u) wredA[wave * 2u + (lane >> 4)] = s;
    if (lane == 8u || lane == 24u) wredB[wave * 2u + (lane >> 4)] = s;
  }
  __syncthreads();
  if (tid == 0) {
    float sa = 0.f, sb = 0.f;
    for (int i = 0; i < 16; ++i) { sa += wredA[i]; sb += wredB[i]; }
    partA[blk] = sa;
    partB[blk] = sb;
  }
}

__global__ __launch_bounds__(TPB) void crit_final(const float* __restrict__ partA,
                                                  const float* __restrict__ partB,
                                                  const float* __restrict__ w2,
                                                  float* __restrict__ out,
                                                  int nblk, int nBatch) {
  __shared__ float ra[TPB], rb[TPB];
  const int t = threadIdx.x;
  float a = 0.f, bb = 0.f;
  for (int i = t; i < nblk; i += TPB) { a += partA[i]; bb += partB[i]; }
  ra[t] = a; rb[t] = bb;
  __syncthreads();
  for (int s = TPB / 2; s > 0; s >>= 1) {
    if (t < s) { ra[t] += ra[t + s]; rb[t] += rb[t + s]; }
    __syncthreads();
  }
  if (t == 0) {
    // singular values of 6x6 W: Jacobi eigensolve of A = W^T W (f64, single lane)
    double A[6][6];
    for (int i = 0; i < 6; ++i)
      for (int j = 0; j < 6; ++j) {
        double s = 0.0;
        for (int k = 0; k < 6; ++k) s += (double)w2[k * 6 + i] * (double)w2[k * 6 + j];
        A[i][j] = s;
      }
    for (int sweep = 0; sweep < 30; ++sweep)
      for (int p = 0; p < 5; ++p)
        for (int q = p + 1; q < 6; ++q) {
          const double apq = A[p][q];
          if (fabs(apq) < 1e-300) continue;
          const double tau = (A[q][q] - A[p][p]) / (2.0 * apq);
          const double tt = (tau >= 0.0 ? 1.0 : -1.0) / (fabs(tau) + sqrt(1.0 + tau * tau));
          const double c = 1.0 / sqrt(1.0 + tt * tt);
          const double sn = tt * c;
          for (int k = 0; k < 6; ++k) {
            const double kp = A[k][p], kq = A[k][q];
            A[k][p] = c * kp - sn * kq;
            A[k][q] = sn * kp + c * kq;
          }
          for (int k = 0; k < 6; ++k) {
            const double pk = A[p][k], qk = A[q][k];
            A[p][k] = c * pk - sn * qk;
            A[q][k] = sn * pk + c * qk;
          }
        }
    double svd = 0.0;
    for (int i = 0; i < 6; ++i) {
      const double e = A[i][i];
      svd += fabs(sqrt(e > 0.0 ? e : 0.0) - 1.0);
    }
    const double nB = (double)nBatch;
    const double loss = (double)ra[0] / (nB * 54.0)
                      + 0.1 * (double)rb[0] / (nB * 6.0)
                      + 0.01 * (svd / 6.0);
    out[0] = (float)loss;
  }
}

extern "C" void kernel_launch(void* const* d_in, const int* in_sizes, int n_in,
                              void* d_out, int out_size, void* d_ws, size_t ws_size,
                              hipStream_t stream) {
  (void)n_in; (void)out_size; (void)ws_size;
  const float* pred = (const float*)d_in[0];
  const float* tar  = (const float*)d_in[1];
  const float* glp  = (const float*)d_in[2];
  const float* w2   = (const float*)d_in[3];
  const int nBatch = in_sizes[2] / 6;          // pr_glpose is (B, 6)
  const int nblk = (nBatch + IPB - 1) / IPB;   // 1024 for B = 262144
  float* partA = (float*)d_ws;
  float* partB = partA + nblk;
  crit_main<<<nblk, TPB, 0, stream>>>(pred, tar, glp, partA, partB, nBatch);
  crit_final<<<1, TPB, 0, stream>>>(partA, partB, w2, (float*)d_out, nblk, nBatch);
}